// LSHAttention_6528350290160
// MI455X (gfx1250) — compile-verified
//
#include <hip/hip_runtime.h>
#include <hip/hip_bf16.h>

// ---------------------------------------------------------------------------
// LSH attention (Reformer-style) for MI455X / gfx1250, wave32, WMMA f32.
// B=8, S=4096, D=64, H=8 hashes, 64 buckets of 64 tokens.
// ---------------------------------------------------------------------------

#define B_    8
#define S_    4096
#define D_    64
#define H_    8
#define NBK   64          // buckets per hash round
#define NBLK  512         // H_ * NBK blocks per batch in sorted space
#define RW    256         // H_ * NBK/2 rotation columns

typedef float v2f __attribute__((ext_vector_type(2)));
typedef float v8f __attribute__((ext_vector_type(8)));

// D = A(16x4) * B(4x16) + C, fp32 WMMA (ISA V_WMMA_F32_16X16X4_F32).
// Per ISA layouts: A reg0/reg1 hold K = 2*(lane>>4) + {0,1} for row M=lane&15.
// B mirrors with N = lane&15. C/D: reg r holds M = r + 8*(lane>>4), N = lane&15.
__device__ __forceinline__ v8f wmma_f32_k4(v2f a, v2f b, v8f c) {
    return __builtin_amdgcn_wmma_f32_16x16x4_f32(
        /*neg_a=*/false, a, /*neg_b=*/false, b,
        /*c_mod=*/(short)0, c, /*reuse_a=*/false, /*reuse_b=*/false);
}

// ---------------------------------------------------------------------------
// Kernel 1: hash.  rotated = qk_chunk[64,64] @ rot[64,256]; bucket = argmax
// over concat([r, -r]) per (token, hash).  One WG per 64-token chunk.
// ---------------------------------------------------------------------------
__global__ __launch_bounds__(256)
void lsh_hash_kernel(const float* __restrict__ qk, const float* __restrict__ rot,
                     unsigned short* __restrict__ bucket_ws, int* __restrict__ bkts_out)
{
    __shared__ float qk_s[64][65];        // padded: stride 65 kills bank aliasing
    __shared__ float rot_s[64 * RW];      // row-major [k][n]
    __shared__ float proj_s[64 * 260];    // [token][256] padded to 260

    const int tid = threadIdx.x;
    const int b  = blockIdx.x >> 6;
    const int s0 = (blockIdx.x & 63) << 6;

    __builtin_prefetch(rot, 0, 0);        // global_prefetch_b8 (speculative, L2)

    // stage qk chunk (64x64 f32) via float4
    {
        const float4* src = (const float4*)(qk + ((size_t)b * S_ + s0) * D_);
        for (int i = tid; i < 64 * 16; i += 256) {
            float4 val = src[i];
            int r = i >> 4, c = (i & 15) << 2;
            qk_s[r][c + 0] = val.x; qk_s[r][c + 1] = val.y;
            qk_s[r][c + 2] = val.z; qk_s[r][c + 3] = val.w;
        }
    }
    // stage rotations (64x256 f32), identical for every block -> L2 resident
    {
        const float4* src = (const float4*)rot;
        float4* dst = (float4*)rot_s;
        for (int i = tid; i < 64 * RW / 4; i += 256) dst[i] = src[i];
    }
    __syncthreads();

    const int lane = tid & 31, wv = tid >> 5;
    const int ml = lane & 15, hi = lane >> 4;

    // 4 (M) x 16 (N) tiles of 16x16; 8 waves -> 8 tiles each; K=64 in steps of 4
    for (int j = 0; j < 8; ++j) {
        const int tt = wv * 8 + j;
        const int mt = tt >> 4, nt = tt & 15;
        const int m = mt * 16 + ml, n = nt * 16 + ml;
        v8f c = {};
        for (int k0 = 0; k0 < 64; k0 += 4) {
            const int ka = k0 + 2 * hi;
            v2f a, bf;
            a.x  = qk_s[m][ka];          a.y  = qk_s[m][ka + 1];
            bf.x = rot_s[ka * RW + n];   bf.y = rot_s[(ka + 1) * RW + n];
            c = wmma_f32_k4(a, bf, c);
        }
        const int mbase = mt * 16 + hi * 8;
#pragma unroll
        for (int r = 0; r < 8; ++r) proj_s[(mbase + r) * 260 + n] = c[r];
    }
    __syncthreads();

    // argmax over concat([r, -r]) (64 candidates); first-max-wins like jnp.argmax
    for (int p = tid; p < 512; p += 256) {
        const int t = p >> 3, h = p & 7;
        const float* row = &proj_s[t * 260 + h * 32];
        float best = -3.4e38f; int bi = 0;
#pragma unroll
        for (int i = 0; i < 32; ++i) { float x =  row[i]; if (x > best) { best = x; bi = i; } }
#pragma unroll
        for (int i = 0; i < 32; ++i) { float x = -row[i]; if (x > best) { best = x; bi = 32 + i; } }
        const size_t gi = ((size_t)b * H_ + h) * S_ + (s0 + t);
        bucket_ws[gi] = (unsigned short)bi;       // local id for counting sort
        bkts_out[gi]  = bi + h * NBK;             // reference's offset bucket output
    }
}

// ---------------------------------------------------------------------------
// Kernel 2: stable counting sort per (b,h): 4096 tokens into 64 buckets,
// stable by position.  One WG (64 threads) per (b,h); thread j owns bucket j.
// ---------------------------------------------------------------------------
__global__ __launch_bounds__(64)
void lsh_sort_kernel(const unsigned short* __restrict__ bucket_ws,
                     unsigned short* __restrict__ sorted_ws)
{
    __shared__ unsigned char lbk[S_];
    __shared__ int counts[NBK];
    __shared__ int offs[NBK];

    const int tid = threadIdx.x;
    const size_t base = (size_t)blockIdx.x * S_;     // blockIdx = b*H + h

    counts[tid] = 0;
    __syncthreads();
    for (int t = tid; t < S_; t += 64) {
        int bk = bucket_ws[base + t];
        lbk[t] = (unsigned char)bk;
        atomicAdd(&counts[bk], 1);
    }
    __syncthreads();
    if (tid == 0) {
        int acc = 0;
        for (int j = 0; j < NBK; ++j) { offs[j] = acc; acc += counts[j]; }
    }
    __syncthreads();
    {   // stable scatter: bucket j scans tokens in position order
        const unsigned char j = (unsigned char)tid;
        int pos = offs[tid];
        for (int t = 0; t < S_; ++t)
            if (lbk[t] == j) sorted_ws[base + pos++] = (unsigned short)t;
    }
}

// ---------------------------------------------------------------------------
// Kernel 3: per-block attention.  One WG per (b, sorted block u in [0,512)).
// Gathers bq/bk/bv, normalizes keys, WMMA dots, self-mask (keep only equal
// positions, as written in the reference), softmax+LSE, WMMA bo, then
// scatter-unsorts directly to o_ws/logits_ws[b][h][t].
// ---------------------------------------------------------------------------
__global__ __launch_bounds__(256)
void lsh_attn_kernel(const float* __restrict__ qk, const float* __restrict__ v,
                     const unsigned short* __restrict__ sorted_ws,
                     float* __restrict__ o_ws, float* __restrict__ logits_ws)
{
    __shared__ float bq_s[64][65];
    __shared__ float bk_s[128][65];
    __shared__ float bv_s[128][65];
    __shared__ float p_s[64 * 132];       // dots -> probs, padded to 132
    __shared__ int   tq[64];
    __shared__ int   tk[128];

    const int tid = threadIdx.x;
    const int b  = blockIdx.x >> 9;
    const int u  = blockIdx.x & 511;
    const int up = (u + 511) & 511;       // look_one_back wraps over all 512 blocks
    const int h  = u >> 6;
    const size_t sbase = (size_t)b * (H_ * S_);   // sorted index is flat h*S+pos

    if (tid < 64)       tk[tid] = tq[tid] = sorted_ws[sbase + u  * 64 + tid];
    else if (tid < 128) tk[tid]           = sorted_ws[sbase + up * 64 + (tid - 64)];
    __syncthreads();

    // gather queries (64 rows) — one (row, quarter) task per thread
    {
        const int r = tid >> 2, q = tid & 3;
        const float4* src = (const float4*)(qk + ((size_t)b * S_ + tq[r]) * D_);
#pragma unroll
        for (int j = 0; j < 4; ++j) {
            float4 val = src[q * 4 + j];
            int c = q * 16 + j * 4;
            bq_s[r][c] = val.x; bq_s[r][c + 1] = val.y;
            bq_s[r][c + 2] = val.z; bq_s[r][c + 3] = val.w;
        }
    }
    // gather keys + values (128 rows each)
    for (int task = tid; task < 512; task += 256) {
        const int r = task >> 2, q = task & 3;
        const int t = tk[r];
        const float4* sk = (const float4*)(qk + ((size_t)b * S_ + t) * D_);
        const float4* sv = (const float4*)(v  + ((size_t)b * S_ + t) * D_);
#pragma unroll
        for (int j = 0; j < 4; ++j) {
            float4 kv = sk[q * 4 + j];
            float4 vv = sv[q * 4 + j];
            int c = q * 16 + j * 4;
            bk_s[r][c] = kv.x; bk_s[r][c + 1] = kv.y;
            bk_s[r][c + 2] = kv.z; bk_s[r][c + 3] = kv.w;
            bv_s[r][c] = vv.x; bv_s[r][c + 1] = vv.y;
            bv_s[r][c + 2] = vv.z; bv_s[r][c + 3] = vv.w;
        }
    }
    __syncthreads();

    // unit-length keys: bk *= rsqrt(sum(bk^2) + 1e-6)
    for (int r = tid; r < 128; r += 256) {
        float ss = 0.f;
#pragma unroll
        for (int c = 0; c < 64; ++c) { float x = bk_s[r][c]; ss += x * x; }
        const float inv = rsqrtf(ss + 1e-6f);
#pragma unroll
        for (int c = 0; c < 64; ++c) bk_s[r][c] *= inv;
    }
    __syncthreads();

    const int lane = tid & 31, wv = tid >> 5;
    const int ml = lane & 15, hi = lane >> 4;

    // dots[64,128] = bq @ bk^T * D^-0.5 ; 4x8 tiles, 4 per wave, K=64
    for (int j = 0; j < 4; ++j) {
        const int tt = wv * 4 + j;
        const int mt = tt >> 3, nt = tt & 7;
        const int m = mt * 16 + ml, n = nt * 16 + ml;
        v8f c = {};
        for (int k0 = 0; k0 < 64; k0 += 4) {
            const int ka = k0 + 2 * hi;
            v2f a, bf;
            a.x  = bq_s[m][ka]; a.y  = bq_s[m][ka + 1];
            bf.x = bk_s[n][ka]; bf.y = bk_s[n][ka + 1];   // B[k][n] = bk[n][k]
            c = wmma_f32_k4(a, bf, c);
        }
        const int mbase = mt * 16 + hi * 8;
#pragma unroll
        for (int r = 0; r < 8; ++r) {
            const int mm = mbase + r;
            float val = c[r] * 0.125f;                    // D^-0.5 = 1/8
            val = (tq[mm] == tk[n]) ? val : -1e5f;        // reference's self-mask
            p_s[mm * 132 + n] = val;
        }
    }
    __syncthreads();

    // row softmax over 128 keys; emit LSE (unsorted to logits_ws directly)
    if (tid < 64) {
        float mx = -3.4e38f;
        for (int jj = 0; jj < 128; ++jj) mx = fmaxf(mx, p_s[tid * 132 + jj]);
        float sum = 0.f;
        for (int jj = 0; jj < 128; ++jj) sum += expf(p_s[tid * 132 + jj] - mx);
        const float lse = mx + logf(sum);
        for (int jj = 0; jj < 128; ++jj)
            p_s[tid * 132 + jj] = expf(p_s[tid * 132 + jj] - lse);
        logits_ws[((size_t)b * H_ + h) * S_ + tq[tid]] = lse;
    }
    __syncthreads();

    // bo[64,64] = probs @ bv ; 4x4 tiles, 2 per wave, K=128; scatter-unsort
    const size_t obase = ((size_t)b * H_ + h) * S_;
    for (int j = 0; j < 2; ++j) {
        const int tt = wv * 2 + j;
        const int mt = tt >> 2, nt = tt & 3;
        const int m = mt * 16 + ml, n = nt * 16 + ml;
        v8f c = {};
        for (int k0 = 0; k0 < 128; k0 += 4) {
            const int ka = k0 + 2 * hi;
            v2f a, bf;
            a.x  = p_s[m * 132 + ka]; a.y  = p_s[m * 132 + ka + 1];
            bf.x = bv_s[ka][n];       bf.y = bv_s[ka + 1][n];
            c = wmma_f32_k4(a, bf, c);
        }
        const int mbase = mt * 16 + hi * 8;
#pragma unroll
        for (int r = 0; r < 8; ++r) {
            const int mm = mbase + r;
            o_ws[(obase + tq[mm]) * D_ + n] = c[r];
        }
    }
}

// ---------------------------------------------------------------------------
// Kernel 4: combine over hash rounds: w_h = softmax_h(lse_h); out = sum w_h*o_h
// ---------------------------------------------------------------------------
__global__ __launch_bounds__(256)
void lsh_combine_kernel(const float* __restrict__ o_ws,
                        const float* __restrict__ logits_ws,
                        float* __restrict__ out)
{
    const int idx = blockIdx.x * 256 + threadIdx.x;   // [0, B*S)
    const int b = idx >> 12, t = idx & 4095;

    float w[H_];
    float mx = -3.4e38f;
#pragma unroll
    for (int h = 0; h < H_; ++h) {
        w[h] = logits_ws[((size_t)b * H_ + h) * S_ + t];
        mx = fmaxf(mx, w[h]);
    }
    float sum = 0.f;
#pragma unroll
    for (int h = 0; h < H_; ++h) { w[h] = expf(w[h] - mx); sum += w[h]; }
    const float inv = 1.f / sum;
#pragma unroll
    for (int h = 0; h < H_; ++h) w[h] *= inv;    // == exp(l - logsumexp(l))

    for (int d = 0; d < D_; d += 4) {
        float4 acc = {0.f, 0.f, 0.f, 0.f};
#pragma unroll
        for (int h = 0; h < H_; ++h) {
            const float4 ov =
                *(const float4*)&o_ws[(((size_t)b * H_ + h) * S_ + t) * D_ + d];
            acc.x += ov.x * w[h]; acc.y += ov.y * w[h];
            acc.z += ov.z * w[h]; acc.w += ov.w * w[h];
        }
        *(float4*)&out[((size_t)b * S_ + t) * D_ + d] = acc;
    }
}

// ---------------------------------------------------------------------------
// Host launcher.
//   inputs:  qk [B,S,D] f32, v [B,S,D] f32, rotations [1,D,H,32] f32
//   outputs: out [B,S,D] f32 then buckets [B,H*S] int32, concatenated in d_out
//   workspace: buckets u16 (512 KB) | sorted u16 (512 KB) | logits f32 (1 MB)
//              | o_ws f32 (64 MB)   — total ~66 MB
// ---------------------------------------------------------------------------
extern "C" void kernel_launch(void* const* d_in, const int* in_sizes, int n_in,
                              void* d_out, int out_size, void* d_ws, size_t ws_size,
                              hipStream_t stream) {
    const float* qk  = (const float*)d_in[0];
    const float* v   = (const float*)d_in[1];
    const float* rot = (const float*)d_in[2];

    float* out = (float*)d_out;
    int*   bkts_out = (int*)((float*)d_out + (size_t)B_ * S_ * D_);

    unsigned short* bucket_ws = (unsigned short*)d_ws;
    unsigned short* sorted_ws = bucket_ws + (size_t)B_ * H_ * S_;
    float* logits_ws = (float*)((char*)d_ws + 2u * (size_t)B_ * H_ * S_ * sizeof(unsigned short));
    float* o_ws      = logits_ws + (size_t)B_ * H_ * S_;

    lsh_hash_kernel   <<<B_ * (S_ / 64), 256, 0, stream>>>(qk, rot, bucket_ws, bkts_out);
    lsh_sort_kernel   <<<B_ * H_,         64, 0, stream>>>(bucket_ws, sorted_ws);
    lsh_attn_kernel   <<<B_ * NBLK,      256, 0, stream>>>(qk, v, sorted_ws, o_ws, logits_ws);
    lsh_combine_kernel<<<(B_ * S_) / 256, 256, 0, stream>>>(o_ws, logits_ws, out);
}